// BSQ_962072674971
// MI455X (gfx1250) — compile-verified
//
#include <hip/hip_runtime.h>
#include <hip/hip_bf16.h>

// Fused BSQ forward for MI455X (gfx1250, wave32).
//   z   = x @ W_enc^T + b_enc     (N=65536 tokens, D=512, C=16)
//   q   = sign(z)                 (L2-normalize dropped: positive scale keeps sign)
//   out = q @ W_dec^T + b_dec
// Memory-bound: 268 MB @ 23.3 TB/s ~= 11.5 us floor vs 2.1 GFLOP fp32.
// x tiles + both weight matrices are staged to LDS by the Tensor Data Mover
// (tensor_load_to_lds, TENSORcnt double-buffered); GEMMs use V_WMMA_F32_16X16X4_F32.

typedef float v2f __attribute__((ext_vector_type(2)));
typedef float v8f __attribute__((ext_vector_type(8)));
typedef unsigned int u32x4 __attribute__((ext_vector_type(4)));
typedef int i32x4 __attribute__((ext_vector_type(4)));
typedef int i32x8 __attribute__((ext_vector_type(8)));

constexpr int DIMS   = 512;
constexpr int WAVES  = 8;            // 256-thread blocks
constexpr int CHUNK  = 128;          // x K-chunk (floats per row)
// TDM pad: interval 128 dw + 4 dw  -> LDS row stride 132 (132%64=4, conflict-free)
constexpr int XBUF_STRIDE = 132;
constexpr int XBUF_FLOATS = 16 * XBUF_STRIDE;            // 2112 floats = 8448 B
// TDM pad: interval 256 dw + 2 dw  -> row stride 516, +2 shift for k>=256
constexpr int WENC_STRIDE = 516;
constexpr int WENC_FLOATS = 16 * WENC_STRIDE;            // 8256 floats
// TDM pad: interval 16 dw + 2 dw   -> exactly row stride 18 (divides the row)
constexpr int WDEC_STRIDE = 18;
constexpr int QT_STRIDE   = 18;

// --- Tensor Data Mover: 2D tile load, D# per CDNA5 ISA ---
__device__ __forceinline__ void tdm_load_2d(
    unsigned lds_off_bytes, const void* gptr,
    unsigned tile0, unsigned tile1,                       // tile dims (elements)
    unsigned long long tdim0, unsigned long long tdim1,   // tensor dims (OOB)
    unsigned long long stride0,                           // dim0 stride (elements)
    unsigned pad_interval_code, unsigned pad_amount_code) // 2^(i+1) dw, (a+1) dw
{
    const unsigned long long ga = (unsigned long long)(uintptr_t)gptr;
    u32x4 g0;
    g0.x = 1u;                                   // count=1 user descriptor
    g0.y = lds_off_bytes;                        // lds_addr [63:32]
    g0.z = (unsigned)ga;                         // global_addr [95:64]
    g0.w = (unsigned)(ga >> 32) | (2u << 30);    // global_addr[56:32] | type=2
    i32x8 g1;
    g1[0] = (int)((2u << 16)                     // data_size = 4 bytes
                | (1u << 20)                     // pad_enable
                | (pad_interval_code << 22)
                | (pad_amount_code << 25));      // workgroup_mask = 0
    g1[1] = (int)((unsigned)(tdim0 & 0xFFFFu) << 16);
    g1[2] = (int)(((tdim0 >> 16) & 0xFFFFu) | ((tdim1 & 0xFFFFu) << 16));
    g1[3] = (int)(((tdim1 >> 16) & 0xFFFFu) | (tile0 << 16));
    g1[4] = (int)(tile1 & 0xFFFFu);              // tile_dim2 = 0
    g1[5] = (int)(unsigned)(stride0 & 0xFFFFFFFFull);
    g1[6] = (int)(unsigned)((stride0 >> 32) & 0xFFFFull); // dim1 stride = 0 (2D)
    g1[7] = 0;
    const i32x4 z4 = {0, 0, 0, 0};
#if defined(__clang_major__) && (__clang_major__ >= 23)
    const i32x8 z8 = {0, 0, 0, 0, 0, 0, 0, 0};
    __builtin_amdgcn_tensor_load_to_lds(g0, g1, z4, z4, z8, 0);
#else
    __builtin_amdgcn_tensor_load_to_lds(g0, g1, z4, z4, 0);
#endif
}

__device__ __forceinline__ unsigned lds_off(const void* p) {
    return (unsigned)(uintptr_t)p;               // low 32 bits = LDS byte offset
}

__global__ __launch_bounds__(256) void bsq_fused_kernel(
    const float* __restrict__ x,
    const float* __restrict__ W_enc,
    const float* __restrict__ b_enc,
    const float* __restrict__ W_dec,
    const float* __restrict__ b_dec,
    float* __restrict__ out)
{
    __shared__ float lds_wenc[WENC_FLOATS];                  //  33.0 KB
    __shared__ float lds_wdec[DIMS * WDEC_STRIDE];           //  36.9 KB
    __shared__ float lds_x[WAVES * 2 * XBUF_FLOATS];         // 132.0 KB
    __shared__ float lds_q[WAVES][16 * QT_STRIDE];           //   9.2 KB

    const int tid  = threadIdx.x;
    const int lane = tid & 31;
    const int wave = tid >> 5;
    const int n16  = lane & 15;   // M/N index within the 16x16 fragment
    const int hi   = lane >> 4;   // lane half selects K-pair / row-half
    const int kOff = hi << 1;

    const int tile = (blockIdx.x * WAVES + wave) << 4;       // 16-token tile base

    // ---- TDM: stage weights (once per block) and first two x chunks -------
    if (wave == 0)       // W_enc 16x512, pad 256dw+2 -> stride 516
        tdm_load_2d(lds_off(lds_wenc), W_enc, DIMS, 16, DIMS, 16, DIMS, 7u, 1u);
    else if (wave == 1)  // W_dec 512x16, pad 16dw+2 -> stride 18
        tdm_load_2d(lds_off(lds_wdec), W_dec, 16, DIMS, 16, DIMS, 16, 3u, 1u);

    float* xbuf0 = &lds_x[(wave * 2 + 0) * XBUF_FLOATS];
    float* xbuf1 = &lds_x[(wave * 2 + 1) * XBUF_FLOATS];
    const float* xg = x + (size_t)tile * DIMS;
    // x chunk c: 16 rows x 128 floats at column c*128, pad 128dw+4 -> stride 132
    tdm_load_2d(lds_off(xbuf0), xg + 0 * CHUNK, CHUNK, 16, DIMS, 1u << 20, DIMS, 6u, 3u);
    tdm_load_2d(lds_off(xbuf1), xg + 1 * CHUNK, CHUNK, 16, DIMS, 1u << 20, DIMS, 6u, 3u);

    if (wave < 2) __builtin_amdgcn_s_wait_tensorcnt(2);      // weights landed
    __syncthreads();                                         // visible block-wide

    // ================= encode: Z(16x16) = X(16x512) * W_enc^T ================
    // A-frag (x) / B-frag (W_enc) per-lane pattern: row = lane%16,
    // cols = kb + 2*(lane/16) + {0,1} -> one 8B-aligned float2 from LDS.
    v8f acc0 = {};
    v8f acc1 = {};
    #pragma unroll
    for (int c = 0; c < 4; ++c) {
        // builtin requires a literal immediate -> constant in each arm
        if (c < 3) __builtin_amdgcn_s_wait_tensorcnt(1);     // chunk c resident
        else       __builtin_amdgcn_s_wait_tensorcnt(0);
        asm volatile("" ::: "memory");
        const float* xb = (c & 1 ? xbuf1 : xbuf0) + n16 * XBUF_STRIDE + kOff;
        const float* wb = &lds_wenc[n16 * WENC_STRIDE + c * CHUNK + (c >= 2 ? 2 : 0) + kOff];
        #pragma unroll 4
        for (int kb = 0; kb < CHUNK; kb += 8) {
            v2f a0 = *(const v2f*)(xb + kb);
            v2f b0 = *(const v2f*)(wb + kb);
            acc0 = __builtin_amdgcn_wmma_f32_16x16x4_f32(
                false, a0, false, b0, (short)0, acc0, false, false);
            v2f a1 = *(const v2f*)(xb + kb + 4);
            v2f b1 = *(const v2f*)(wb + kb + 4);
            acc1 = __builtin_amdgcn_wmma_f32_16x16x4_f32(
                false, a1, false, b1, (short)0, acc1, false, false);
        }
        if (c < 2) {     // prefetch chunk c+2 into the buffer just consumed
            asm volatile("" ::: "memory");
            tdm_load_2d(lds_off(c & 1 ? xbuf1 : xbuf0), xg + (c + 2) * CHUNK,
                        CHUNK, 16, DIMS, 1u << 20, DIMS, 6u, 3u);
        }
    }

    // bias + straight-through sign (forward value only)
    const float be = b_enc[n16];
    float q[8];
    #pragma unroll
    for (int r = 0; r < 8; ++r) {
        const float z = acc0[r] + acc1[r] + be;
        q[r] = (z >= 0.0f) ? 1.0f : -1.0f;
    }

    // transpose Q from C/D layout to A layout via per-wave LDS (same-wave LDS
    // ops execute in order; no workgroup barrier required)
    float* qt = lds_q[wave];
    #pragma unroll
    for (int r = 0; r < 8; ++r)
        qt[(r + (hi << 3)) * QT_STRIDE + n16] = q[r];

    v2f aq[4];
    #pragma unroll
    for (int k2 = 0; k2 < 4; ++k2)
        aq[k2] = *(const v2f*)&qt[n16 * QT_STRIDE + (k2 << 2) + kOff];

    // ============== decode: Out(16x512) = Q(16x16) * W_dec^T + b_dec =========
    float* orow = out + ((size_t)(tile + (hi << 3))) * DIMS + n16;
    #pragma unroll 2
    for (int nb = 0; nb < DIMS; nb += 16) {
        const float bd = b_dec[nb + n16];
        v8f cc = { bd, bd, bd, bd, bd, bd, bd, bd };
        #pragma unroll
        for (int k2 = 0; k2 < 4; ++k2) {
            v2f bw = *(const v2f*)&lds_wdec[(nb + n16) * WDEC_STRIDE + (k2 << 2) + kOff];
            cc = __builtin_amdgcn_wmma_f32_16x16x4_f32(
                false, aq[k2], false, bw, (short)0, cc, false, false);
        }
        #pragma unroll
        for (int r = 0; r < 8; ++r)
            orow[(size_t)r * DIMS + nb] = cc[r];   // 2x 64B contiguous segments
    }
}

extern "C" void kernel_launch(void* const* d_in, const int* in_sizes, int n_in,
                              void* d_out, int out_size, void* d_ws, size_t ws_size,
                              hipStream_t stream) {
    const float* x     = (const float*)d_in[0];
    const float* W_enc = (const float*)d_in[1];
    const float* b_enc = (const float*)d_in[2];
    const float* W_dec = (const float*)d_in[3];
    const float* b_dec = (const float*)d_in[4];
    float* out = (float*)d_out;

    // 65536 tokens / (8 waves * 16 tokens) = 512 blocks
    dim3 grid(512), block(256);
    hipLaunchKernelGGL(bsq_fused_kernel, grid, block, 0, stream,
                       x, W_enc, b_enc, W_dec, b_dec, out);
}